// Quantum_5884105195996
// MI455X (gfx1250) — compile-verified
//
#include <hip/hip_runtime.h>

// 8-qubit batched state-vector simulator for MI455X (gfx1250, wave32).
// One wave per batch element, full 256-amplitude state in VGPRs
// (8 complex amps / lane). Cross-lane gate pairing via shfl_xor.
// Readout (probs x signs GEMM) via chained V_WMMA_F32_16X16X4_F32.

#define NQ 8
#define NSTATES 256
#define WAVES_PER_BLOCK 16
#define THREADS_PER_BLOCK (WAVES_PER_BLOCK * 32)
#define WCOLS 72 /* (N_LAYERS_CRZ + N) * N = 9*8 */

typedef float v2f __attribute__((ext_vector_type(2)));
typedef float v8f __attribute__((ext_vector_type(8)));

__global__ __launch_bounds__(THREADS_PER_BLOCK)
void quantum_circuit_kernel(const float* __restrict__ X,
                            const float* __restrict__ W,
                            float* __restrict__ out,
                            int B)
{
    __shared__ float probs[WAVES_PER_BLOCK][NSTATES]; // 16 KB

    const int lane  = threadIdx.x & 31;
    const int wave  = threadIdx.x >> 5;
    const int bBase = blockIdx.x * WAVES_PER_BLOCK;
    const int b     = bBase + wave;
    const int bc    = (b < B) ? b : 0;
    const float* xb = X + (size_t)bc * (2 * NQ);

    // state: s = lane*8 + k ; qubit q <-> bit (7-q) of s
    float ar[8], ai[8];
    #pragma unroll
    for (int k = 0; k < 8; ++k) { ar[k] = 0.f; ai[k] = 0.f; }
    if (lane == 0) ar[0] = 1.f; // |00000000>

    float pr[8], pi_[8]; // partner amplitudes (target-bit flipped)

    // ---- encoding layer: per qubit q: H, RZ(X[2q]), RY(X[2q+1]) ----
    #pragma unroll
    for (int q = 0; q < NQ; ++q) {
        const int bp = 7 - q;

        // H : new = (partner + sign*mine)/sqrt2
        #pragma unroll
        for (int k = 0; k < 8; ++k) {
            if (bp < 3) { pr[k] = ar[k ^ (1 << bp)]; pi_[k] = ai[k ^ (1 << bp)]; }
            else        { pr[k] = __shfl_xor(ar[k], 1 << (bp - 3), 32);
                          pi_[k] = __shfl_xor(ai[k], 1 << (bp - 3), 32); }
        }
        #pragma unroll
        for (int k = 0; k < 8; ++k) {
            const int s = (lane << 3) | k;
            const float sg = ((s >> bp) & 1) ? -1.f : 1.f;
            ar[k] = (pr[k]  + sg * ar[k]) * 0.70710678f;
            ai[k] = (pi_[k] + sg * ai[k]) * 0.70710678f;
        }

        // RZ(theta): a *= exp(-+ i*theta/2) by bit
        {
            const float h = 0.5f * xb[2 * q];
            float sh, ch; __sincosf(h, &sh, &ch);
            #pragma unroll
            for (int k = 0; k < 8; ++k) {
                const int s = (lane << 3) | k;
                const float ps = ((s >> bp) & 1) ? sh : -sh; // imag(phase)
                const float nr = ar[k] * ch - ai[k] * ps;
                const float ni = ar[k] * ps + ai[k] * ch;
                ar[k] = nr; ai[k] = ni;
            }
        }

        // RY(theta): new = cos*mine + (bit? +sin : -sin)*partner (real mtx)
        {
            const float h = 0.5f * xb[2 * q + 1];
            float sy, cy; __sincosf(h, &sy, &cy);
            #pragma unroll
            for (int k = 0; k < 8; ++k) {
                if (bp < 3) { pr[k] = ar[k ^ (1 << bp)]; pi_[k] = ai[k ^ (1 << bp)]; }
                else        { pr[k] = __shfl_xor(ar[k], 1 << (bp - 3), 32);
                              pi_[k] = __shfl_xor(ai[k], 1 << (bp - 3), 32); }
            }
            #pragma unroll
            for (int k = 0; k < 8; ++k) {
                const int s = (lane << 3) | k;
                const float t = ((s >> bp) & 1) ? sy : -sy;
                ar[k] = cy * ar[k] + t * pr[k];
                ai[k] = cy * ai[k] + t * pi_[k];
            }
        }
    }

    // ---- CRZ ring: control i, target (i+1)%8, angle weights[0,i] (diag) ----
    #pragma unroll
    for (int i = 0; i < NQ; ++i) {
        const int t  = (i + 1) & 7;
        const int cb = 7 - i;
        const int tb = 7 - t;
        const float h = 0.5f * W[i];
        float sh, ch; __sincosf(h, &sh, &ch);
        #pragma unroll
        for (int k = 0; k < 8; ++k) {
            const int s = (lane << 3) | k;
            const bool ctrl = (s >> cb) & 1;
            const float ps = ((s >> tb) & 1) ? sh : -sh;
            const float nr = ar[k] * ch - ai[k] * ps;
            const float ni = ar[k] * ps + ai[k] * ch;
            ar[k] = ctrl ? nr : ar[k];
            ai[k] = ctrl ? ni : ai[k];
        }
    }

    // ---- CRX all-to-all: if ctrl: new = cos*mine - i*sin*partner ----
    #pragma unroll
    for (int i = 0; i < NQ; ++i) {
        #pragma unroll
        for (int j = 0; j < NQ; ++j) {
            if (j == i) continue;
            const int cb = 7 - i;
            const int tb = 7 - j;
            const float h = 0.5f * W[(1 + i) * WCOLS + j];
            float sx, cx; __sincosf(h, &sx, &cx);
            #pragma unroll
            for (int k = 0; k < 8; ++k) {
                if (tb < 3) { pr[k] = ar[k ^ (1 << tb)]; pi_[k] = ai[k ^ (1 << tb)]; }
                else        { pr[k] = __shfl_xor(ar[k], 1 << (tb - 3), 32);
                              pi_[k] = __shfl_xor(ai[k], 1 << (tb - 3), 32); }
            }
            #pragma unroll
            for (int k = 0; k < 8; ++k) {
                const int s = (lane << 3) | k;
                const bool ctrl = (s >> cb) & 1;
                const float nr = cx * ar[k] + sx * pi_[k]; // -i*s*(pr+i*pi)
                const float ni = cx * ai[k] - sx * pr[k];
                ar[k] = ctrl ? nr : ar[k];
                ai[k] = ctrl ? ni : ai[k];
            }
        }
    }

    // ---- probabilities to LDS: P[wave][s] ----
    #pragma unroll
    for (int k = 0; k < 8; ++k)
        probs[wave][(lane << 3) | k] = ar[k] * ar[k] + ai[k] * ai[k];

    __syncthreads();

    // ---- readout: Out(16x16) = P(16x256) * Sign(256x16), cols>=8 zero.
    // Chained V_WMMA_F32_16X16X4_F32, K advances 4 per step (64 steps).
    // A (16x4 f32): lanes 0-15 hold M=0-15 K={0,1}, lanes 16-31 K={2,3}.
    // B (4x16 f32): lanes 0-15 hold N=0-15 K={0,1}, lanes 16-31 K={2,3}.
    if (threadIdx.x < 32) { // wave 0: EXEC all-1s (WMMA requirement)
        const int L     = threadIdx.x;
        const int mn    = L & 15;   // M for A rows, N for B/C/D cols
        const int khalf = L >> 4;
        v8f acc = {0.f, 0.f, 0.f, 0.f, 0.f, 0.f, 0.f, 0.f};
        #pragma unroll 4
        for (int kk = 0; kk < 64; ++kk) {
            const int s0 = kk * 4 + 2 * khalf;
            v2f A; A.x = probs[mn][s0]; A.y = probs[mn][s0 + 1];
            float b0 = 0.f, b1 = 0.f;
            if (mn < 8) { // Sign[s][q] = (bit_q(s) ? -1 : +1), q = mn
                b0 = (((s0)     >> (7 - mn)) & 1) ? -1.f : 1.f;
                b1 = (((s0 + 1) >> (7 - mn)) & 1) ? -1.f : 1.f;
            }
            v2f Bv; Bv.x = b0; Bv.y = b1;
            acc = __builtin_amdgcn_wmma_f32_16x16x4_f32(
                false, A, false, Bv, (short)0, acc, false, false);
        }
        // C/D layout: VGPR r -> M = r + 8*khalf, N = mn
        #pragma unroll
        for (int r = 0; r < 8; ++r) {
            const int m  = r + 8 * khalf;
            const int bb = bBase + m;
            if (mn < 8 && bb < B) out[bb * NQ + mn] = acc[r];
        }
    }
}

extern "C" void kernel_launch(void* const* d_in, const int* in_sizes, int n_in,
                              void* d_out, int out_size, void* d_ws, size_t ws_size,
                              hipStream_t stream) {
    (void)n_in; (void)out_size; (void)d_ws; (void)ws_size;
    const float* X = (const float*)d_in[0];   // (B, 16) f32
    const float* W = (const float*)d_in[1];   // (9, 72) f32
    float* out = (float*)d_out;               // (B, 8) f32
    const int B = in_sizes[0] / (2 * NQ);
    const int grid = (B + WAVES_PER_BLOCK - 1) / WAVES_PER_BLOCK;
    quantum_circuit_kernel<<<grid, THREADS_PER_BLOCK, 0, stream>>>(X, W, out, B);
}